// AttnDecoder_12790412608088
// MI455X (gfx1250) — compile-verified
//
#include <hip/hip_runtime.h>
#include <math.h>

#define Bsz 32
#define Hd  128
#define NHh 4
#define HDd 32
#define Vv  32000
#define Tt  82
#define Ss  82

typedef float v2f __attribute__((ext_vector_type(2)));
typedef float v8f __attribute__((ext_vector_type(8)));

// ---------------- K/V projection of encoder outputs (run once) ----------------
__global__ __launch_bounds__(128) void kv_kernel(
    const float* __restrict__ enc, const float* __restrict__ ipw,
    const float* __restrict__ ipb, float* __restrict__ kbuf,
    float* __restrict__ vbuf)
{
  __shared__ float e[Hd];
  int row = blockIdx.x;          // b*S + s
  int j = threadIdx.x;           // 0..127
  e[j] = enc[row * Hd + j];
  __syncthreads();
  const float* wk = ipw + (size_t)(Hd + j) * Hd;      // wk row j
  const float* wv = ipw + (size_t)(2 * Hd + j) * Hd;  // wv row j
  float accK = ipb[Hd + j];
  float accV = ipb[2 * Hd + j];
#pragma unroll 8
  for (int c = 0; c < Hd; ++c) { accK += e[c] * wk[c]; accV += e[c] * wv[c]; }
  kbuf[row * Hd + j] = accK;
  vbuf[row * Hd + j] = accV;
}

// ---------------- per-step attention + out_proj + embedding -> x ----------------
__global__ __launch_bounds__(256) void attn_kernel(
    const float* __restrict__ h, const float* __restrict__ kbuf,
    const float* __restrict__ vbuf, const float* __restrict__ ipw,
    const float* __restrict__ ipb, const float* __restrict__ opw,
    const float* __restrict__ opb, const float* __restrict__ emb,
    const int* __restrict__ target, float* __restrict__ xbuf,
    float* __restrict__ attn_out, int t)
{
  __shared__ float sq[Bsz * Hd];   // q, later reused for ctx (16 KB)
  __shared__ float sw[128 * 84];   // per-(b,head) score rows (43 KB)
  int tid = threadIdx.x;

  // q = h @ wq^T + bq
  for (int o = tid; o < Bsz * Hd; o += 256) {
    int b = o >> 7, j = o & 127;
    const float* hr = h + b * Hd;
    const float* wq = ipw + (size_t)j * Hd;
    float acc = ipb[j];
#pragma unroll 8
    for (int c = 0; c < Hd; ++c) acc += hr[c] * wq[c];
    sq[o] = acc;
  }
  __syncthreads();

  const float scale = 0.17677669529663687f;  // 1/sqrt(32)
  if (tid < 128) {                           // one (b, head) per thread
    int b = tid >> 2, n = tid & 3;
    const float* qp = sq + b * Hd + n * HDd;
    float mx = -1e30f;
    for (int s = 0; s < Ss; ++s) {
      const float* kp = kbuf + (size_t)(b * Ss + s) * Hd + n * HDd;
      float acc = 0.f;
#pragma unroll 8
      for (int d = 0; d < HDd; ++d) acc += qp[d] * kp[d];
      acc *= scale;
      sw[tid * 84 + s] = acc;
      mx = fmaxf(mx, acc);
    }
    float sum = 0.f;
    for (int s = 0; s < Ss; ++s) {
      float e = __expf(sw[tid * 84 + s] - mx);
      sw[tid * 84 + s] = e;
      sum += e;
    }
    float inv = 1.f / sum;
    for (int s = 0; s < Ss; ++s) sw[tid * 84 + s] *= inv;
  }
  __syncthreads();

  // attention averaged over heads -> output tensor [B,T,S]
  for (int o = tid; o < Bsz * Ss; o += 256) {
    int b = o / Ss, s = o % Ss;
    float a = 0.25f * (sw[(b * 4 + 0) * 84 + s] + sw[(b * 4 + 1) * 84 + s] +
                       sw[(b * 4 + 2) * 84 + s] + sw[(b * 4 + 3) * 84 + s]);
    attn_out[(size_t)b * Tt * Ss + (size_t)t * Ss + s] = a;
  }
  // ctx = w @ v  (reuse sq region; q no longer needed)
  if (tid < 128) {
    int b = tid >> 2, n = tid & 3;
    for (int d = 0; d < HDd; ++d) {
      float acc = 0.f;
      for (int s = 0; s < Ss; ++s)
        acc += sw[tid * 84 + s] * vbuf[(size_t)(b * Ss + s) * Hd + n * HDd + d];
      sq[b * Hd + n * HDd + d] = acc;
    }
  }
  __syncthreads();

  // out_proj(ctx) -> x[:,128:256]; embedding lookup -> x[:,0:128]
  for (int o = tid; o < Bsz * Hd; o += 256) {
    int b = o >> 7, j = o & 127;
    const float* cr = sq + b * Hd;
    const float* wr = opw + (size_t)j * Hd;
    float acc = opb[j];
#pragma unroll 8
    for (int c = 0; c < Hd; ++c) acc += cr[c] * wr[c];
    xbuf[b * 256 + 128 + j] = acc;
    int tok = (t == 0) ? 0 : target[b * Tt + (t - 1)];
    xbuf[b * 256 + j] = emb[(size_t)tok * Hd + j];
  }
}

// ---------------- GRU cell: h_new from x=[emb,ctx] and h ----------------
__global__ __launch_bounds__(128) void gru_kernel(
    const float* __restrict__ x, const float* __restrict__ h,
    const float* __restrict__ wih, const float* __restrict__ bih,
    const float* __restrict__ whh, const float* __restrict__ bhh,
    float* __restrict__ hnew)
{
  __shared__ float xr[256];
  __shared__ float hr[128];
  int b = blockIdx.x, j = threadIdx.x;
  xr[j] = x[b * 256 + j];
  xr[128 + j] = x[b * 256 + 128 + j];
  hr[j] = h[b * 128 + j];
  __syncthreads();

  float gi_r = bih[j], gi_z = bih[128 + j], gi_n = bih[256 + j];
  const float* wr = wih + (size_t)j * 256;
  const float* wz = wih + (size_t)(128 + j) * 256;
  const float* wn = wih + (size_t)(256 + j) * 256;
#pragma unroll 4
  for (int c = 0; c < 256; ++c) {
    float xv = xr[c];
    gi_r += xv * wr[c]; gi_z += xv * wz[c]; gi_n += xv * wn[c];
  }
  float gh_r = bhh[j], gh_z = bhh[128 + j], gh_n = bhh[256 + j];
  const float* ur = whh + (size_t)j * 128;
  const float* uz = whh + (size_t)(128 + j) * 128;
  const float* un = whh + (size_t)(256 + j) * 128;
#pragma unroll 4
  for (int c = 0; c < 128; ++c) {
    float hv = hr[c];
    gh_r += hv * ur[c]; gh_z += hv * uz[c]; gh_n += hv * un[c];
  }
  float r = 1.f / (1.f + __expf(-(gi_r + gh_r)));
  float z = 1.f / (1.f + __expf(-(gi_z + gh_z)));
  float n = tanhf(gi_n + r * gh_n);
  hnew[b * 128 + j] = (1.f - z) * n + z * hr[j];
}

// ---------------- logits = h_new(32x128) @ out_w^T + out_b via fp32 WMMA ----------------
// Wave: one 16-vocab-column strip, both 16-row batch tiles, K=128 in 32 x WMMA(16x16x4 f32).
__global__ __launch_bounds__(256) void logits_wmma_kernel(
    const float* __restrict__ h, const float* __restrict__ outw,
    const float* __restrict__ outb, float* __restrict__ logits)
{
  int lane = threadIdx.x & 31;
  int wave = threadIdx.x >> 5;
  int nbase = blockIdx.x * 128 + wave * 16;  // 250 blocks * 128 = 32000 exactly
  int half = lane >> 4;                      // K sub-pair selector
  int l16 = lane & 15;

  const float* arow0 = h + (size_t)l16 * Hd;         // batch rows 0..15
  const float* arow1 = h + (size_t)(16 + l16) * Hd;  // batch rows 16..31
  const float* brow  = outw + (size_t)(nbase + l16) * Hd;  // vocab row (B^T column)

  v8f c0 = {};
  v8f c1 = {};
#pragma unroll
  for (int kk = 0; kk < Hd; kk += 4) {
    int ko = kk + half * 2;
    v2f a0 = *(const v2f*)(arow0 + ko);   // A frag: lane holds m=l16, k=ko..ko+1
    v2f a1 = *(const v2f*)(arow1 + ko);
    v2f bb = *(const v2f*)(brow + ko);    // B frag: lane holds n=l16, k=ko..ko+1
    c0 = __builtin_amdgcn_wmma_f32_16x16x4_f32(false, a0, false, bb, (short)0, c0,
                                               false, false);
    c1 = __builtin_amdgcn_wmma_f32_16x16x4_f32(false, a1, false, bb, (short)0, c1,
                                               false, false);
  }
  int v = nbase + l16;
  float bias = outb[v];
#pragma unroll
  for (int r = 0; r < 8; ++r) {
    int m = r + 8 * half;                 // D layout: vgpr r -> row r + 8*(lane/16)
    logits[(size_t)m * Vv + v] = c0[r] + bias;
    logits[(size_t)(16 + m) * Vv + v] = c1[r] + bias;
  }
}

// ---------------- log_softmax over vocab, one block per batch row ----------------
__global__ __launch_bounds__(256) void lsm_kernel(
    const float* __restrict__ logits, float* __restrict__ out, int t)
{
  __shared__ float red[256];
  int b = blockIdx.x, tid = threadIdx.x;
  const float* row = logits + (size_t)b * Vv;

  float mx = -1e30f;
  for (int v = tid; v < Vv; v += 256) mx = fmaxf(mx, row[v]);
  red[tid] = mx;
  __syncthreads();
  for (int s2 = 128; s2 > 0; s2 >>= 1) {
    if (tid < s2) red[tid] = fmaxf(red[tid], red[tid + s2]);
    __syncthreads();
  }
  mx = red[0];
  __syncthreads();

  float sum = 0.f;
  for (int v = tid; v < Vv; v += 256) sum += __expf(row[v] - mx);
  red[tid] = sum;
  __syncthreads();
  for (int s2 = 128; s2 > 0; s2 >>= 1) {
    if (tid < s2) red[tid] += red[tid + s2];
    __syncthreads();
  }
  float lse = mx + __logf(red[0]);

  float* orow = out + (size_t)b * Tt * Vv + (size_t)t * Vv;
  for (int v = tid; v < Vv; v += 256) orow[v] = row[v] - lse;
}

extern "C" void kernel_launch(void* const* d_in, const int* in_sizes, int n_in,
                              void* d_out, int out_size, void* d_ws, size_t ws_size,
                              hipStream_t stream) {
  (void)in_sizes; (void)n_in; (void)out_size; (void)ws_size;
  const float* enc  = (const float*)d_in[0];   // [B,S,H]
  const float* ehid = (const float*)d_in[1];   // [1,B,H]
  const int*   tgt  = (const int*)d_in[2];     // [B,T]
  const float* emb  = (const float*)d_in[3];   // [V,H]
  const float* ipw  = (const float*)d_in[4];   // [3H,H]
  const float* ipb  = (const float*)d_in[5];   // [3H]
  const float* opw  = (const float*)d_in[6];   // [H,H]
  const float* opb  = (const float*)d_in[7];   // [H]
  const float* wih  = (const float*)d_in[8];   // [3H,2H]
  const float* bih  = (const float*)d_in[9];   // [3H]
  const float* whh  = (const float*)d_in[10];  // [3H,H]
  const float* bhh  = (const float*)d_in[11];  // [3H]
  const float* outw = (const float*)d_in[12];  // [V,H]
  const float* outb = (const float*)d_in[13];  // [V]

  float* out = (float*)d_out;
  float* ws  = (float*)d_ws;
  // workspace layout (floats)
  float* kbuf   = ws;                  // 32*82*128 = 335872
  float* vbuf   = kbuf + 335872;       // 335872
  float* xbuf   = vbuf + 335872;       // 32*256 = 8192
  float* h0     = xbuf + 8192;         // 4096
  float* h1     = h0 + 4096;           // 4096
  float* logits = h1 + 4096;           // 32*32000 = 1024000  (~4 MB, L2-resident)

  float* logp = out;                                 // [B,T,V]
  float* hfin = out + (size_t)Bsz * Tt * Vv;         // [1,B,H]
  float* attn = hfin + Bsz * Hd;                     // [B,T,S]

  kv_kernel<<<Bsz * Ss, 128, 0, stream>>>(enc, ipw, ipb, kbuf, vbuf);
  hipMemcpyAsync(h0, ehid, Bsz * Hd * sizeof(float), hipMemcpyDeviceToDevice, stream);

  for (int t = 0; t < Tt; ++t) {
    float* hc = (t & 1) ? h1 : h0;
    float* hn = (t & 1) ? h0 : h1;
    attn_kernel<<<1, 256, 0, stream>>>(hc, kbuf, vbuf, ipw, ipb, opw, opb, emb,
                                       tgt, xbuf, attn, t);
    gru_kernel<<<Bsz, 128, 0, stream>>>(xbuf, hc, wih, bih, whh, bhh, hn);
    logits_wmma_kernel<<<Vv / 128, 256, 0, stream>>>(hn, outw, outb, logits);
    lsm_kernel<<<Bsz, 256, 0, stream>>>(logits, logp, t);
  }
  // T=82 is even, so final h lives in h0
  hipMemcpyAsync(hfin, h0, Bsz * Hd * sizeof(float), hipMemcpyDeviceToDevice, stream);
}